// Model_79594333929934
// MI455X (gfx1250) — compile-verified
//
#include <hip/hip_runtime.h>

// CDNA5 (gfx1250) fused LSTM-stack implementation.
// One kernel launch per direction-layer; each wave owns 16 batch rows and
// runs the whole T=512 recurrent scan with weights resident in VGPRs and the
// per-step GEMMs done with V_WMMA_F32_16X16X4_F32 (hidden=16 -> exact tile fit).

typedef __attribute__((ext_vector_type(2))) float v2f;
typedef __attribute__((ext_vector_type(8))) float v8f;

#define DEVINL __device__ __forceinline__

DEVINL v8f wmma4(v2f a, v2f b, v8f c) {
  // D = A(16x4 f32) * B(4x16 f32) + C(16x16 f32)
  return __builtin_amdgcn_wmma_f32_16x16x4_f32(false, a, false, b, (short)0, c,
                                               false, false);
}

DEVINL float sigf(float x) { return 1.0f / (1.0f + __expf(-x)); }
DEVINL float tanh_fast(float x) { return 2.0f * sigf(2.0f * x) - 1.0f; }

struct LstmArgs {
  const float* x;         // (B, T, IN) fp32
  float* y;               // (B, T, OF) fp32; dir d writes features [d*OUT, (d+1)*OUT)
  const float* Wih[2];    // (64, IN)
  const float* Whh[2];    // (64, OUT)
  const float* bih[2];    // (64)
  const float* bhh[2];    // (64)
  const float* Whr[2];    // (OUT=proj, 16)
  const float* h0[2];     // (B, OUT) or null
  const float* c0[2];     // (B, 16) or null
  float* hT[2];           // (B, OUT) or null
  float* cT[2];           // (B, 16) or null
};

// Block = 64 threads (2 waves); grid = (4, D). Wave handles 16 batch rows.
template <int IN, int OUT, bool BIAS, bool PROJ>
__launch_bounds__(64) __global__ void lstm_layer(LstmArgs a, int T, int OF) {
  constexpr int HID = 16;
  constexpr int INC = (IN + 3) / 4;    // K-chunks of 4 over input features
  constexpr int OUTC = (OUT + 3) / 4;  // K-chunks of 4 over recurrent features

  const int d = blockIdx.y;            // direction
  const int lane = threadIdx.x & 31;
  const int wave = threadIdx.x >> 5;
  const int half = lane >> 4;          // lane half selects K sub-pair
  const int l16 = lane & 15;
  const int b0 = blockIdx.x * 32 + wave * 16;  // first batch row of this wave
  const int kA = 2 * half;

  __shared__ float lds_tile[2][16][17];
  float(*tile)[17] = lds_tile[wave];   // private per-wave bounce tile

  const float* __restrict__ Wih = a.Wih[d];
  const float* __restrict__ Whh = a.Whh[d];

  // ---- Preload weights into B-matrix layout (resident across the scan) ----
  // Assumed B layout (mirrors documented A/C pattern): per K-chunk kc,
  // VGPR.x holds K=4kc+2*half, VGPR.y holds K=4kc+2*half+1, N = lane&15.
  v2f wihB[INC][4];
#pragma unroll
  for (int kc = 0; kc < INC; ++kc) {
    const int k0 = 4 * kc + kA;
#pragma unroll
    for (int n = 0; n < 4; ++n) {
      const int g = 16 * n + l16;  // gate row (0..63): n=0:i 1:f 2:g 3:o
      v2f b;
      b.x = (k0 + 0 < IN) ? Wih[g * IN + k0 + 0] : 0.0f;
      b.y = (k0 + 1 < IN) ? Wih[g * IN + k0 + 1] : 0.0f;
      wihB[kc][n] = b;
    }
  }
  v2f whhB[OUTC][4];
#pragma unroll
  for (int kc = 0; kc < OUTC; ++kc) {
    const int k0 = 4 * kc + kA;
#pragma unroll
    for (int n = 0; n < 4; ++n) {
      const int g = 16 * n + l16;
      v2f b;
      b.x = (k0 + 0 < OUT) ? Whh[g * OUT + k0 + 0] : 0.0f;
      b.y = (k0 + 1 < OUT) ? Whh[g * OUT + k0 + 1] : 0.0f;
      whhB[kc][n] = b;
    }
  }
  v2f whrB[4];
#pragma unroll
  for (int kc = 0; kc < 4; ++kc) { whrB[kc].x = 0.0f; whrB[kc].y = 0.0f; }
  if constexpr (PROJ) {
    const float* __restrict__ Whr = a.Whr[d];  // (OUT, 16); B = Whr^T, N-pad->16
#pragma unroll
    for (int kc = 0; kc < 4; ++kc) {
      const int k0 = 4 * kc + kA;
      v2f b;
      b.x = (l16 < OUT) ? Whr[l16 * HID + k0 + 0] : 0.0f;
      b.y = (l16 < OUT) ? Whr[l16 * HID + k0 + 1] : 0.0f;
      whrB[kc] = b;
    }
  }

  float bias_v[4];
#pragma unroll
  for (int n = 0; n < 4; ++n) {
    if constexpr (BIAS)
      bias_v[n] = a.bih[d][16 * n + l16] + a.bhh[d][16 * n + l16];
    else
      bias_v[n] = 0.0f;
  }

  // ---- Initial state ----
  v2f hA[OUTC];  // h in A-matrix layout
  if (a.h0[d]) {
    const float* h0 = a.h0[d];
#pragma unroll
    for (int kc = 0; kc < OUTC; ++kc) {
      const int k0 = 4 * kc + kA;
      v2f h;
      h.x = (k0 + 0 < OUT) ? h0[(b0 + l16) * OUT + k0 + 0] : 0.0f;
      h.y = (k0 + 1 < OUT) ? h0[(b0 + l16) * OUT + k0 + 1] : 0.0f;
      hA[kc] = h;
    }
  } else {
#pragma unroll
    for (int kc = 0; kc < OUTC; ++kc) { hA[kc].x = 0.0f; hA[kc].y = 0.0f; }
  }
  v8f cst;  // cell state, C/D layout: row = v + 8*half, col = l16
  if (a.c0[d]) {
#pragma unroll
    for (int v = 0; v < 8; ++v)
      cst[v] = a.c0[d][(b0 + v + 8 * half) * HID + l16];
  } else {
#pragma unroll
    for (int v = 0; v < 8; ++v) cst[v] = 0.0f;
  }

  float hC[8];  // last h in C/D layout (row v+8*half, col l16)
#pragma unroll
  for (int v = 0; v < 8; ++v) hC[v] = 0.0f;

  // ---- Recurrent scan (direction 1 runs time reversed) ----
  int t = d ? (T - 1) : 0;
  const int dt = d ? -1 : 1;
  for (int s = 0; s < T; ++s, t += dt) {
    v8f acc[4];
#pragma unroll
    for (int n = 0; n < 4; ++n)
#pragma unroll
      for (int v = 0; v < 8; ++v) acc[n][v] = bias_v[n];

    // g += x_t @ Wih^T   (A = 16-row batch tile of x_t)
    const float* xrow = a.x + (b0 + l16) * T * IN + t * IN;
#pragma unroll
    for (int kc = 0; kc < INC; ++kc) {
      const int k0 = 4 * kc + kA;
      v2f xa;
      xa.x = (k0 + 0 < IN) ? xrow[k0 + 0] : 0.0f;
      xa.y = (k0 + 1 < IN) ? xrow[k0 + 1] : 0.0f;
#pragma unroll
      for (int n = 0; n < 4; ++n) acc[n] = wmma4(xa, wihB[kc][n], acc[n]);
    }
    // g += h @ Whh^T
#pragma unroll
    for (int kc = 0; kc < OUTC; ++kc)
#pragma unroll
      for (int n = 0; n < 4; ++n) acc[n] = wmma4(hA[kc], whhB[kc][n], acc[n]);

    // Gates: tiles n = 0:i 1:f 2:g 3:o are element-aligned in C/D layout.
    float hf[8];
#pragma unroll
    for (int v = 0; v < 8; ++v) {
      const float iv = sigf(acc[0][v]);
      const float fv = sigf(acc[1][v]);
      const float gv = tanh_fast(acc[2][v]);
      const float ov = sigf(acc[3][v]);
      const float cv = fv * cst[v] + iv * gv;
      cst[v] = cv;
      hf[v] = ov * tanh_fast(cv);
    }

    if constexpr (PROJ) {
      // h = hf @ Whr^T : bounce hf C->A through LDS, then one WMMA column-tile.
      __builtin_amdgcn_wave_barrier();
#pragma unroll
      for (int v = 0; v < 8; ++v) tile[v + 8 * half][l16] = hf[v];
      __builtin_amdgcn_wave_barrier();
      v8f p;
#pragma unroll
      for (int v = 0; v < 8; ++v) p[v] = 0.0f;
#pragma unroll
      for (int kc = 0; kc < 4; ++kc) {
        const int k0 = 4 * kc + kA;
        v2f ha;
        ha.x = tile[l16][k0 + 0];
        ha.y = tile[l16][k0 + 1];
        p = wmma4(ha, whrB[kc], p);  // cols >= OUT are exactly 0 (zero-pad B)
      }
#pragma unroll
      for (int v = 0; v < 8; ++v) hC[v] = p[v];
    } else {
#pragma unroll
      for (int v = 0; v < 8; ++v) hC[v] = hf[v];
    }

    // Bounce h C->A for the next step's recurrent GEMM.
    __builtin_amdgcn_wave_barrier();
#pragma unroll
    for (int v = 0; v < 8; ++v) tile[v + 8 * half][l16] = hC[v];
    __builtin_amdgcn_wave_barrier();
#pragma unroll
    for (int kc = 0; kc < OUTC; ++kc) {
      const int k0 = 4 * kc + kA;
      hA[kc].x = tile[l16][k0 + 0];
      hA[kc].y = tile[l16][k0 + 1];
    }
    __builtin_amdgcn_wave_barrier();

    // Emit y[b, t, d*OUT + j]
#pragma unroll
    for (int v = 0; v < 8; ++v) {
      const int m = v + 8 * half;
      if (l16 < OUT)
        a.y[(b0 + m) * T * OF + t * OF + d * OUT + l16] = hC[v];
    }
  }

  // Final carried state for the next stack (state chaining s0 -> s1).
  if (a.hT[d]) {
#pragma unroll
    for (int v = 0; v < 8; ++v) {
      const int m = v + 8 * half;
      if (l16 < OUT) a.hT[d][(b0 + m) * OUT + l16] = hC[v];
    }
  }
  if (a.cT[d]) {
#pragma unroll
    for (int v = 0; v < 8; ++v) {
      const int m = v + 8 * half;
      a.cT[d][(b0 + m) * HID + l16] = cst[v];
    }
  }
}

// ---------------------------------------------------------------------------
// Host side
// ---------------------------------------------------------------------------

struct W5 {
  const float *Wih, *Whh, *bih, *bhh, *Whr;
};

template <int IN, int OUT, bool BIAS, bool PROJ>
static void launch_one(const float* x, float* y, int OF, int D, const W5* w,
                       const float* sHin, const float* sCin, float* sHout,
                       float* sCout, int li, hipStream_t st) {
  LstmArgs a{};
  a.x = x;
  a.y = y;
  for (int dd = 0; dd < D; ++dd) {
    a.Wih[dd] = w[dd].Wih;
    a.Whh[dd] = w[dd].Whh;
    a.bih[dd] = w[dd].bih;
    a.bhh[dd] = w[dd].bhh;
    a.Whr[dd] = w[dd].Whr;
    const int slot = li * 2 + dd;  // jax stacking order: layer-major, dir-minor
    a.h0[dd] = sHin ? sHin + slot * 128 * OUT : nullptr;
    a.c0[dd] = sCin ? sCin + slot * 128 * 16 : nullptr;
    a.hT[dd] = sHout ? sHout + slot * 128 * OUT : nullptr;
    a.cT[dd] = sCout ? sCout + slot * 128 * 16 : nullptr;
  }
  dim3 grid(4, D);
  lstm_layer<IN, OUT, BIAS, PROJ><<<grid, 64, 0, st>>>(a, 512, OF);
}

extern "C" void kernel_launch(void* const* d_in, const int* in_sizes, int n_in,
                              void* d_out, int out_size, void* d_ws,
                              size_t ws_size, hipStream_t stream) {
  (void)in_sizes; (void)n_in; (void)out_size; (void)ws_size;

  const float* x_in = (const float*)d_in[0];  // (128, 512, 32)
  const float* y_in = (const float*)d_in[1];  // (128, 512, 25)

  // Flatten order: per spec (l00..l14), per layer, per dir:
  // Wih, Whh, [bih, bhh], [Whr]; None dropped; no key leaves.
  int idx = 2;
  auto nf = [&]() { return (const float*)d_in[idx++]; };
  W5 p00[1], p01[3], p02[8], p03[8], p04[8];
  W5 p10[1], p11[3], p12[8], p13[8], p14[8];
  auto read_spec = [&](W5* dst, int Lc, int Dc, bool bias, bool proj) {
    for (int l = 0; l < Lc; ++l)
      for (int dd = 0; dd < Dc; ++dd) {
        W5 w{};
        w.Wih = nf();
        w.Whh = nf();
        if (bias) { w.bih = nf(); w.bhh = nf(); }
        if (proj) { w.Whr = nf(); }
        dst[l * Dc + dd] = w;
      }
  };
  read_spec(p00, 1, 1, true, false);
  read_spec(p01, 3, 1, false, false);
  read_spec(p02, 4, 2, true, true);
  read_spec(p03, 4, 2, true, true);
  read_spec(p04, 4, 2, true, true);
  read_spec(p10, 1, 1, true, false);
  read_spec(p11, 3, 1, false, false);
  read_spec(p12, 4, 2, true, true);
  read_spec(p13, 4, 2, true, true);
  read_spec(p14, 4, 2, true, true);

  const int B = 128, T = 512;
  const int N20 = B * T * 20;
  float* out = (float*)d_out;
  float* Rx2 = out + 0 * N20;
  float* Rx3 = out + 1 * N20;
  float* Rx4 = out + 2 * N20;
  float* Ry2 = out + 3 * N20;
  float* Ry3 = out + 4 * N20;
  float* Ry4 = out + 5 * N20;

  // Workspace: one (B,T,20) ping buffer + state arrays (h:(8,B,10), c:(8,B,16)).
  float* wsf = (float*)d_ws;
  float* bufC = wsf;                    // N20 floats
  float* sH0 = wsf + N20;               // 8*128*10
  float* sC0 = sH0 + 8 * 128 * 10;      // 8*128*16
  float* sH1 = sC0 + 8 * 128 * 16;
  float* sC1 = sH1 + 8 * 128 * 10;

  // ---------------- x branch ----------------
  // l00 / l01 (16-feat intermediates scribble into not-yet-final out regions)
  launch_one<32, 16, true, false>(x_in, Rx2, 16, 1, p00, 0, 0, 0, 0, 0, stream);
  launch_one<16, 16, false, false>(Rx2, Rx3, 16, 1, &p01[0], 0, 0, 0, 0, 0, stream);
  launch_one<16, 16, false, false>(Rx3, Rx2, 16, 1, &p01[1], 0, 0, 0, 0, 0, stream);
  launch_one<16, 16, false, false>(Rx2, Rx3, 16, 1, &p01[2], 0, 0, 0, 0, 0, stream);
  // l02 (bidir, proj) -> final x2 in Rx2, state -> s0
  launch_one<16, 10, true, true>(Rx3, bufC, 20, 2, &p02[0], 0, 0, sH0, sC0, 0, stream);
  launch_one<20, 10, true, true>(bufC, Rx2, 20, 2, &p02[2], 0, 0, sH0, sC0, 1, stream);
  launch_one<20, 10, true, true>(Rx2, bufC, 20, 2, &p02[4], 0, 0, sH0, sC0, 2, stream);
  launch_one<20, 10, true, true>(bufC, Rx2, 20, 2, &p02[6], 0, 0, sH0, sC0, 3, stream);
  // l03 (state in s0, out s1) -> final x3 in Rx3
  launch_one<20, 10, true, true>(Rx2, bufC, 20, 2, &p03[0], sH0, sC0, sH1, sC1, 0, stream);
  launch_one<20, 10, true, true>(bufC, Rx3, 20, 2, &p03[2], sH0, sC0, sH1, sC1, 1, stream);
  launch_one<20, 10, true, true>(Rx3, bufC, 20, 2, &p03[4], sH0, sC0, sH1, sC1, 2, stream);
  launch_one<20, 10, true, true>(bufC, Rx3, 20, 2, &p03[6], sH0, sC0, sH1, sC1, 3, stream);
  // l04 (state in s1) -> final x4 in Rx4
  launch_one<20, 10, true, true>(Rx3, bufC, 20, 2, &p04[0], sH1, sC1, 0, 0, 0, stream);
  launch_one<20, 10, true, true>(bufC, Rx4, 20, 2, &p04[2], sH1, sC1, 0, 0, 1, stream);
  launch_one<20, 10, true, true>(Rx4, bufC, 20, 2, &p04[4], sH1, sC1, 0, 0, 2, stream);
  launch_one<20, 10, true, true>(bufC, Rx4, 20, 2, &p04[6], sH1, sC1, 0, 0, 3, stream);

  // ---------------- y branch ----------------
  launch_one<25, 16, true, false>(y_in, Ry2, 16, 1, p10, 0, 0, 0, 0, 0, stream);
  launch_one<16, 16, false, false>(Ry2, Ry3, 16, 1, &p11[0], 0, 0, 0, 0, 0, stream);
  launch_one<16, 16, false, false>(Ry3, Ry2, 16, 1, &p11[1], 0, 0, 0, 0, 0, stream);
  launch_one<16, 16, false, false>(Ry2, Ry3, 16, 1, &p11[2], 0, 0, 0, 0, 0, stream);
  launch_one<16, 10, true, true>(Ry3, bufC, 20, 2, &p12[0], 0, 0, sH0, sC0, 0, stream);
  launch_one<20, 10, true, true>(bufC, Ry2, 20, 2, &p12[2], 0, 0, sH0, sC0, 1, stream);
  launch_one<20, 10, true, true>(Ry2, bufC, 20, 2, &p12[4], 0, 0, sH0, sC0, 2, stream);
  launch_one<20, 10, true, true>(bufC, Ry2, 20, 2, &p12[6], 0, 0, sH0, sC0, 3, stream);
  launch_one<20, 10, true, true>(Ry2, bufC, 20, 2, &p13[0], sH0, sC0, sH1, sC1, 0, stream);
  launch_one<20, 10, true, true>(bufC, Ry3, 20, 2, &p13[2], sH0, sC0, sH1, sC1, 1, stream);
  launch_one<20, 10, true, true>(Ry3, bufC, 20, 2, &p13[4], sH0, sC0, sH1, sC1, 2, stream);
  launch_one<20, 10, true, true>(bufC, Ry3, 20, 2, &p13[6], sH0, sC0, sH1, sC1, 3, stream);
  launch_one<20, 10, true, true>(Ry3, bufC, 20, 2, &p14[0], sH1, sC1, 0, 0, 0, stream);
  launch_one<20, 10, true, true>(bufC, Ry4, 20, 2, &p14[2], sH1, sC1, 0, 0, 1, stream);
  launch_one<20, 10, true, true>(Ry4, bufC, 20, 2, &p14[4], sH1, sC1, 0, 0, 2, stream);
  launch_one<20, 10, true, true>(bufC, Ry4, 20, 2, &p14[6], sH1, sC1, 0, 0, 3, stream);

  (void)T;
}